// yolov2_loss_36103495090633
// MI455X (gfx1250) — compile-verified
//
#include <hip/hip_runtime.h>
#include <hip/hip_bf16.h>
#include <math.h>

// Problem constants (from reference)
#define B_  64
#define A_  5
#define C_  80
#define H_  52
#define W_  52
#define T_  32
#define HW_  (H_ * W_)
#define AHW_ (A_ * HW_)
#define N_   (B_ * AHW_)          // 865,280 = 3380 * 256 exactly

#define L_COORD 5.0f
#define L_OBJ   1.0f
#define L_NOOBJ 0.5f

typedef __attribute__((ext_vector_type(2))) float v2f;
typedef __attribute__((ext_vector_type(8))) float v8f;

__device__ __forceinline__ float softplus_f(float x) {
    // stable: max(x,0) + log1p(exp(-|x|)); matches jax.nn.softplus forward
    return fmaxf(x, 0.0f) + log1pf(expf(-fabsf(x)));
}

__device__ __forceinline__ float sigmoid_f(float x) {
    return 1.0f / (1.0f + expf(-x));
}

__device__ __forceinline__ float iou_with(float px1, float py1, float px2, float py2,
                                          float parea,
                                          float tx, float ty, float tw, float th,
                                          int gix, int giy) {
    float cx  = tx + (float)gix;
    float cy  = ty + (float)giy;
    float gx1 = cx - tw * 0.5f;
    float gy1 = cy - th * 0.5f;
    float gx2 = gx1 + tw;
    float gy2 = gy1 + th;
    float dx = fmaxf(fminf(px2, gx2) - fmaxf(px1, gx1), 0.0f);
    float dy = fmaxf(fminf(py2, gy2) - fmaxf(py1, gy1), 0.0f);
    float inter = dx * dy;
    return inter / (parea + tw * th - inter);   // unguarded, same as reference
}

// Wave32 full-lane sum via V_WMMA_F32_16X16X4_F32.
// A: VGPR0 = per-lane value (K=0 lanes 0-15, K=2 lanes 16-31), VGPR1 = 0 (K=1/3).
// B: all ones -> D[m,n] = v_m + v_{m+16} for every n.
// C/D layout: lane l, VGPR g holds (M = g + (l<16 ? 0 : 8), N = l%16).
// Sum of a lane's 8 D regs = sum r[0..7] (lanes 0-15) or r[8..15] (lanes 16-31);
// one xor-16 shuffle finishes the reduction. Requires EXEC = all ones.
__device__ __forceinline__ float wave_sum_wmma(float v) {
    v2f a; a[0] = v;    a[1] = 0.0f;
    v2f b; b[0] = 1.0f; b[1] = 1.0f;
    v8f c = {};
    c = __builtin_amdgcn_wmma_f32_16x16x4_f32(false, a, false, b,
                                              (short)0, c, false, false);
    float s = c[0] + c[1] + c[2] + c[3] + c[4] + c[5] + c[6] + c[7];
    s += __shfl_xor(s, 16, 32);
    return s;
}

// Kernel 0: zero the 5 output accumulators, precompute per-(b,t) flat keys
// key = tgt_idx_box*HW + tgt_idx_y*W + tgt_idx_x  (unique per batch: perm-based)
__global__ void __launch_bounds__(256)
yolo_init_kernel(const int* __restrict__ tix, const int* __restrict__ tiy,
                 const int* __restrict__ tib, int* __restrict__ keys,
                 float* __restrict__ out) {
    int i = blockIdx.x * blockDim.x + threadIdx.x;
    if (i < 5) out[i] = 0.0f;
    if (i < B_ * T_) keys[i] = tib[i] * HW_ + tiy[i] * W_ + tix[i];
}

// Kernel 1: fused YOLOv2 loss. One thread per (b, a, y, x) cell.
__global__ void __launch_bounds__(256)
yolo_loss_kernel(const float* __restrict__ pred_cls,
                 const float* __restrict__ pred_response,
                 const float* __restrict__ pred_bboxes,
                 const float* __restrict__ tgt_box,
                 const int*   __restrict__ tgt_label,
                 const int*   __restrict__ keys,
                 float* __restrict__ out) {
    int gid = blockIdx.x * blockDim.x + threadIdx.x;

    float l_pos = 0.0f, l_neg = 0.0f, l_cls = 0.0f, l_xy = 0.0f, l_wh = 0.0f;

    if (gid < N_) {
        int b  = gid / AHW_;
        int r  = gid - b * AHW_;        // a*HW + y*W + x  == this cell's key
        int a  = r / HW_;
        int yx = r - a * HW_;
        int y  = yx / W_;
        int x  = yx - y * W_;

        // raw bbox channels: layout (B, A, 4, H, W)
        const float* pbb = pred_bboxes + (((size_t)b * A_ + a) * 4) * HW_ + yx;
        float t0 = pbb[0 * HW_];
        float t1 = pbb[1 * HW_];
        float pw = pbb[2 * HW_];
        float ph = pbb[3 * HW_];

        float px1 = sigmoid_f(t0) + (float)x - pw * 0.5f;
        float py1 = sigmoid_f(t1) + (float)y - ph * 0.5f;
        float px2 = px1 + pw;
        float py2 = py1 + ph;
        float parea = pw * ph;

        // positive-match scan over the 32 targets of this batch
        const int* kb = keys + b * T_;
        int post = -1;
#pragma unroll
        for (int t = 0; t < T_; ++t) {
            if (kb[t] == r) post = t;
        }

        float resp = pred_response[gid];
        const float invB = 1.0f / (float)B_;

        if (post >= 0) {
            const float* tb = tgt_box + ((size_t)b * T_ + post) * 4;
            float tx = tb[0], ty = tb[1], tw = tb[2], th = tb[3];
            int kk  = kb[post];
            int gix = kk % W_;
            int giy = (kk / W_) % H_;

            float iou_sel = iou_with(px1, py1, px2, py2, parea, tx, ty, tw, th, gix, giy);
            l_pos = (softplus_f(resp) - resp * iou_sel) * (L_OBJ * invB);

            // xy: BCE on raw offsets vs tgt xy; wh: squared error
            l_xy = (softplus_f(t0) - t0 * tx + softplus_f(t1) - t1 * ty) * invB;
            float dw = pw - tw, dh = ph - th;
            l_wh = (dw * dw + dh * dh) * (L_COORD * invB);

            // class cross-entropy over C=80 strided logits
            const float* lg = pred_cls + (((size_t)b * A_ + a) * C_) * HW_ + yx;
            float m = -INFINITY;
#pragma unroll 4
            for (int ci = 0; ci < C_; ++ci) m = fmaxf(m, lg[(size_t)ci * HW_]);
            float ssum = 0.0f;
#pragma unroll 4
            for (int ci = 0; ci < C_; ++ci) ssum += expf(lg[(size_t)ci * HW_] - m);
            int lbl = tgt_label[b * T_ + post];
            float lse = m + logf(ssum);
            l_cls = (lse - lg[(size_t)lbl * HW_]) * invB;
        } else {
            // negative: only if IoU with the LAST target of this batch < 0.6
            const float* tbl = tgt_box + ((size_t)b * T_ + (T_ - 1)) * 4;
            int kl  = kb[T_ - 1];
            int gix = kl % W_;
            int giy = (kl / W_) % H_;
            float iou_last = iou_with(px1, py1, px2, py2, parea,
                                      tbl[0], tbl[1], tbl[2], tbl[3], gix, giy);
            if (iou_last < 0.6f) l_neg = softplus_f(resp) * (L_NOOBJ * invB);
        }
    }

    // Reconvergent point: full EXEC. WMMA-based wave32 reductions (one per loss).
    float s0 = wave_sum_wmma(l_pos);
    float s1 = wave_sum_wmma(l_neg);
    float s2 = wave_sum_wmma(l_cls);
    float s3 = wave_sum_wmma(l_xy);
    float s4 = wave_sum_wmma(l_wh);

    if ((threadIdx.x & 31) == 0) {
        atomicAdd(&out[0], s0);
        atomicAdd(&out[1], s1);
        atomicAdd(&out[2], s2);
        atomicAdd(&out[3], s3);
        atomicAdd(&out[4], s4);
    }
}

extern "C" void kernel_launch(void* const* d_in, const int* in_sizes, int n_in,
                              void* d_out, int out_size, void* d_ws, size_t ws_size,
                              hipStream_t stream) {
    const float* pred_cls      = (const float*)d_in[0];
    const float* pred_response = (const float*)d_in[1];
    const float* pred_bboxes   = (const float*)d_in[2];
    const float* tgt_box       = (const float*)d_in[3];
    const int*   tgt_idx_x     = (const int*)d_in[4];
    const int*   tgt_idx_y     = (const int*)d_in[5];
    const int*   tgt_idx_box   = (const int*)d_in[6];
    const int*   tgt_label     = (const int*)d_in[7];

    float* out = (float*)d_out;
    int*   keys = (int*)d_ws;            // B*T ints = 8 KB of workspace

    // Zero outputs + build keys
    yolo_init_kernel<<<(B_ * T_ + 255) / 256, 256, 0, stream>>>(
        tgt_idx_x, tgt_idx_y, tgt_idx_box, keys, out);

    // Fused loss: exactly N_/256 blocks (no ragged tail)
    yolo_loss_kernel<<<N_ / 256, 256, 0, stream>>>(
        pred_cls, pred_response, pred_bboxes, tgt_box, tgt_label, keys, out);
}